// PNA_Irreps_48137993454072
// MI455X (gfx1250) — compile-verified
//
#include <hip/hip_runtime.h>
#include <math.h>

typedef float v2f __attribute__((ext_vector_type(2)));
typedef float v8f __attribute__((ext_vector_type(8)));

#define FEAT 240
#define NIRR 112

// ---------------------------------------------------------------- zero ws
__global__ void zero_f32(float* __restrict__ p, long n) {
  long i = (long)blockIdx.x * blockDim.x + threadIdx.x;
  long stride = (long)gridDim.x * blockDim.x;
  for (; i < n; i += stride) p[i] = 0.0f;
}

// ------------------------------------------------- per-node irrep norm helper
__device__ __forceinline__ float irrep_norm(const float* row, int t) {
  int base, d;
  if (t < 64)      { base = t;                  d = 1; }
  else if (t < 96) { base = 64 + (t - 64) * 3;  d = 3; }
  else             { base = 160 + (t - 96) * 5; d = 5; }
  float s = 0.f;
#pragma unroll 5
  for (int j = 0; j < 5; ++j)
    if (j < d) { float v = row[base + j]; s += v * v; }
  return sqrtf(s);
}

// ---------------------------------------------------------------- pass 1
// one block per node: norms -> exp(+-n) segment sums, x segment sum, counts
__global__ __launch_bounds__(128) void pass1(
    const float* __restrict__ x, const int* __restrict__ index,
    float* __restrict__ z_p, float* __restrict__ z_m,
    float* __restrict__ sum_x, float* __restrict__ cnt, int N) {
  __shared__ float row[FEAT];
  int node = blockIdx.x;
  if (node >= N) return;
  int t = threadIdx.x;
  int g = index[node];
  const float* xr = x + (long)node * FEAT;
  row[t] = xr[t];
  if (t < FEAT - 128) row[t + 128] = xr[t + 128];
  __syncthreads();
  if (t < NIRR) {
    float n = irrep_norm(row, t);
    atomicAdd(&z_p[g * NIRR + t], __expf(n));
    atomicAdd(&z_m[g * NIRR + t], __expf(-n));
  }
  atomicAdd(&sum_x[g * FEAT + t], row[t]);
  if (t < FEAT - 128) atomicAdd(&sum_x[g * FEAT + t + 128], row[t + 128]);
  if (t == 0) atomicAdd(&cnt[g], 1.0f);
}

// ---------------------------------------------------------------- mean
__global__ void finalize_mean(const float* __restrict__ sum_x,
                              const float* __restrict__ cnt,
                              float* __restrict__ mean, int G) {
  int i = blockIdx.x * blockDim.x + threadIdx.x;
  if (i >= G * FEAT) return;
  int g = i / FEAT;
  mean[i] = sum_x[i] / fmaxf(cnt[g], 1.0f);
}

// ---------------------------------------------------------------- pass 2
// recompute norms (x is streaming anyway), apply normalized softmax/softmin
// weights, accumulate weighted pooled sums.
__global__ __launch_bounds__(128) void pass2(
    const float* __restrict__ x, const int* __restrict__ index,
    const float* __restrict__ z_p, const float* __restrict__ z_m,
    float* __restrict__ mx, float* __restrict__ mn, int N) {
  __shared__ float row[FEAT];
  __shared__ float wp[NIRR];
  __shared__ float wm[NIRR];
  int node = blockIdx.x;
  if (node >= N) return;
  int t = threadIdx.x;
  int g = index[node];
  const float* xr = x + (long)node * FEAT;
  row[t] = xr[t];
  if (t < FEAT - 128) row[t + 128] = xr[t + 128];
  __syncthreads();
  if (t < NIRR) {
    float n = irrep_norm(row, t);
    wp[t] = __expf(n)  / z_p[g * NIRR + t];
    wm[t] = __expf(-n) / z_m[g * NIRR + t];
  }
  __syncthreads();
  {
    int j = t;
    int k = (j < 64) ? j : (j < 160) ? 64 + (j - 64) / 3 : 96 + (j - 160) / 5;
    atomicAdd(&mx[g * FEAT + j], row[j] * wp[k]);
    atomicAdd(&mn[g * FEAT + j], row[j] * wm[k]);
  }
  if (t < FEAT - 128) {
    int j = t + 128;
    int k = (j < 160) ? 64 + (j - 64) / 3 : 96 + (j - 160) / 5;
    atomicAdd(&mx[g * FEAT + j], row[j] * wp[k]);
    atomicAdd(&mn[g * FEAT + j], row[j] * wm[k]);
  }
}

// ---------------------------------------------------------------- WMMA linear
// out[g,o,dd] = sum_i cat[g,i,dd] * W[i,o] / sqrt(3*mul)  (+ b0 for group 0)
// cat rows i in [0,3*mul): pool (i/mul) in {mean,mx,mn} (contiguous, stride
// G*FEAT), irrep instance i%mul. One wave per (graph-tile, out-tile, dd).
// A 16x4 f32 layout: lanes 0-15 hold K=k0,k0+1 (v0,v1); lanes 16-31 K=k0+2,k0+3.
// B 4x16: same K split across lane halves, N = lane&15.
// D 16x16: VGPR r -> M = r + 8*(lane>=16), N = lane&15.
// Out-of-range graph rows load a clamped (valid) address; their D rows are
// simply never stored, so no divergent masking in the K-loop.
template <int GI>
__global__ __launch_bounds__(32) void wmma_linear(
    const float* __restrict__ pools,   // mean | mx | mn, each G*FEAT floats
    const float* __restrict__ W,       // (3*mul, mul) row-major
    const float* __restrict__ b0,      // used only when GI==0
    float* __restrict__ out, int G) {
  constexpr int mul = (GI == 0) ? 64 : (GI == 1) ? 32 : 16;
  constexpr int d   = (GI == 0) ? 1  : (GI == 1) ? 3  : 5;
  constexpr int st  = (GI == 0) ? 0  : (GI == 1) ? 64 : 160;
  constexpr int K3  = 3 * mul;
  constexpr int NT  = mul / 16;        // output-channel tiles

  int bid   = blockIdx.x;
  int gtile = bid / (NT * d);
  int r     = bid % (NT * d);
  int ntile = r / d;
  int dd    = r % d;

  const float scale = rsqrtf((float)K3);
  const long poolstride = (long)G * FEAT;

  int lane = threadIdx.x;
  int half = lane >> 4;
  int lrow = lane & 15;
  int grow = gtile * 16 + lrow;
  int gc   = (grow < G) ? grow : (G - 1);      // address clamp only
  int n    = ntile * 16 + lrow;
  long rbase = (long)gc * FEAT + st + dd;

  v8f c = {0.f, 0.f, 0.f, 0.f, 0.f, 0.f, 0.f, 0.f};
#pragma unroll
  for (int k0 = 0; k0 < K3; k0 += 4) {
    int ka = k0 + half * 2;
    int kb = ka + 1;
    int w0i = ka / mul, i0 = ka % mul;         // mul is a power-of-2 constant
    int w1i = kb / mul, i1 = kb % mul;
    float a0 = pools[w0i * poolstride + rbase + i0 * d];
    float a1 = pools[w1i * poolstride + rbase + i1 * d];
    float bv0 = W[ka * mul + n];
    float bv1 = W[kb * mul + n];
    v2f a = {a0, a1};
    v2f b = {bv0, bv1};
    c = __builtin_amdgcn_wmma_f32_16x16x4_f32(false, a, false, b,
                                              (short)0, c, false, false);
  }

#pragma unroll
  for (int rr = 0; rr < 8; ++rr) {
    int m = rr + half * 8;
    int g = gtile * 16 + m;
    if (g < G) {
      float v = c[rr] * scale;
      if (GI == 0) v += b0[n];
      out[(long)g * FEAT + st + n * d + dd] = v;
    }
  }
}

// ---------------------------------------------------------------- launch
extern "C" void kernel_launch(void* const* d_in, const int* in_sizes, int n_in,
                              void* d_out, int out_size, void* d_ws, size_t ws_size,
                              hipStream_t stream) {
  const float* x     = (const float*)d_in[0];
  const int*   index = (const int*)d_in[1];
  const float* W0    = (const float*)d_in[3];
  const float* W1    = (const float*)d_in[4];
  const float* W2    = (const float*)d_in[5];
  const float* b0    = (const float*)d_in[6];
  float* out = (float*)d_out;

  int N = in_sizes[1];          // number of nodes
  int G = out_size / FEAT;      // number of graphs (1000)

  float* ws = (float*)d_ws;
  long o = 0;
  float* z_p   = ws + o; o += (long)G * NIRR;
  float* z_m   = ws + o; o += (long)G * NIRR;
  float* sum_x = ws + o; o += (long)G * FEAT;
  float* cnt   = ws + o; o += G;
  float* mean  = ws + o; o += (long)G * FEAT;  // mean,mx,mn contiguous
  float* mx    = ws + o; o += (long)G * FEAT;
  float* mn    = ws + o; o += (long)G * FEAT;
  long nzero = o;   // ~1.19M floats (~4.7 MB) — lives in L2

  zero_f32<<<1024, 256, 0, stream>>>(ws, nzero);
  pass1<<<N, 128, 0, stream>>>(x, index, z_p, z_m, sum_x, cnt, N);
  finalize_mean<<<(G * FEAT + 255) / 256, 256, 0, stream>>>(sum_x, cnt, mean, G);
  pass2<<<N, 128, 0, stream>>>(x, index, z_p, z_m, mx, mn, N);

  int gtiles = (G + 15) / 16;
  // group0: 4 ntiles * d=1; group1: 2 ntiles * d=3; group2: 1 ntile * d=5
  wmma_linear<0><<<gtiles * 4, 32, 0, stream>>>(mean, W0, b0, out, G);
  wmma_linear<1><<<gtiles * 6, 32, 0, stream>>>(mean, W1, b0, out, G);
  wmma_linear<2><<<gtiles * 5, 32, 0, stream>>>(mean, W2, b0, out, G);
}